// ParallelConvSSM3D_13091060318339
// MI455X (gfx1250) — compile-verified
//
#include <hip/hip_runtime.h>
#include <math.h>

// Problem dims (B=1): x is (T,H,W,C) f32, kernels are (C,3,7,7)
#define TT 16
#define HH 64
#define WW 64
#define CC 128
#define TOT   (TT*HH*WW*CC)   // 8388608 elements
#define STR_T (HH*WW*CC)      // 524288
#define STR_H (WW*CC)         // 8192
#define STR_W (CC)            // 128
#define PI_F 3.14159265358979323846f

typedef __attribute__((ext_vector_type(2))) float v2f;
typedef __attribute__((ext_vector_type(8))) float v8f;

// D(16x16 f32) = A(16x4 f32) * B(4x16 f32) + C  — fp32 matrix pipe, no precision loss
static __device__ __forceinline__ v8f wmma_f32(v2f a, v2f b, v8f c) {
  return __builtin_amdgcn_wmma_f32_16x16x4_f32(false, a, false, b, (short)0, c, false, false);
}

// ---------------------------------------------------------------------------
// Stage 1: forward DFT-16 along T, real input -> complex (re/im planes).
// One wave = one 16-column tile (16 consecutive (h,w,c) points), all 16 rows.
// ---------------------------------------------------------------------------
__global__ void __launch_bounds__(256)
dft16_fwd_real(const float* __restrict__ x, float* __restrict__ ore, float* __restrict__ oim)
{
  __shared__ float Fre[256];
  __shared__ float Fim[256];
  int tid = threadIdx.x;
  {
    int k = tid >> 4, m = tid & 15;
    int km = (k * m) & 15;
    float s, c;
    sincosf(-2.0f * PI_F * (float)km / 16.0f, &s, &c);
    Fre[tid] = c; Fim[tid] = s;
  }
  __syncthreads();

  int lane = tid & 31;
  int tile = blockIdx.x * 8 + (tid >> 5);
  int col  = tile * 16 + (lane & 15);
  int khalf = (lane >> 4) << 1;
  int m = lane & 15;

  v8f zero = {};
  v8f accre = zero, accim = zero;
#pragma unroll
  for (int kk = 0; kk < 4; ++kk) {
    int k0 = kk * 4 + khalf;
    v2f b;   b.x   = x[k0 * STR_T + col];   b.y   = x[(k0 + 1) * STR_T + col];
    v2f are; are.x = Fre[k0 * 16 + m];      are.y = Fre[(k0 + 1) * 16 + m];
    v2f aim; aim.x = Fim[k0 * 16 + m];      aim.y = Fim[(k0 + 1) * 16 + m];
    accre = wmma_f32(are, b, accre);
    accim = wmma_f32(aim, b, accim);
  }
  int rbase = (lane >> 4) * 8;
#pragma unroll
  for (int r = 0; r < 8; ++r) {
    int mm = rbase + r;
    ore[mm * STR_T + col] = accre[r];
    oim[mm * STR_T + col] = accim[r];
  }
}

// ---------------------------------------------------------------------------
// Final stage: inverse DFT-16 along T, complex input -> REAL output only:
// Re(out) = Fre*Xre - Fim*Xim, scaled by 1/(16*64*64).
// ---------------------------------------------------------------------------
__global__ void __launch_bounds__(256)
dft16_inv_real(const float* __restrict__ ire, const float* __restrict__ iim,
               float* __restrict__ out)
{
  __shared__ float Fre[256];
  __shared__ float Fim[256];
  int tid = threadIdx.x;
  {
    int k = tid >> 4, m = tid & 15;
    int km = (k * m) & 15;
    float s, c;
    sincosf(2.0f * PI_F * (float)km / 16.0f, &s, &c);
    Fre[tid] = c; Fim[tid] = s;
  }
  __syncthreads();

  int lane = tid & 31;
  int tile = blockIdx.x * 8 + (tid >> 5);
  int col  = tile * 16 + (lane & 15);
  int khalf = (lane >> 4) << 1;
  int m = lane & 15;

  v8f zero = {};
  v8f acc = zero;
#pragma unroll
  for (int kk = 0; kk < 4; ++kk) {
    int k0 = kk * 4 + khalf;
    v2f bre;  bre.x  = ire[k0 * STR_T + col];  bre.y  = ire[(k0 + 1) * STR_T + col];
    v2f bim;  bim.x  = iim[k0 * STR_T + col];  bim.y  = iim[(k0 + 1) * STR_T + col];
    v2f are;  are.x  = Fre[k0 * 16 + m];       are.y  = Fre[(k0 + 1) * 16 + m];
    v2f aimn; aimn.x = -Fim[k0 * 16 + m];      aimn.y = -Fim[(k0 + 1) * 16 + m];
    acc = wmma_f32(are,  bre, acc);
    acc = wmma_f32(aimn, bim, acc);
  }
  const float sc = 1.0f / (float)(TT * HH * WW);
  int rbase = (lane >> 4) * 8;
#pragma unroll
  for (int r = 0; r < 8; ++r)
    out[(rbase + r) * STR_T + col] = acc[r] * sc;
}

// ---------------------------------------------------------------------------
// Complex<->complex DFT-64 along an axis with given element stride.
// 64x64 DFT matrix in LDS; one wave computes a 64x16 output slab:
//   4 row-tiles x (re,im) accumulators = 8 x v8f, K=64 in 16 WMMA k-chunks.
// Cre = Fre*Xre + (-Fim)*Xim ; Cim = Fre*Xim + Fim*Xre  (4 WMMA per k-chunk/tile)
// ---------------------------------------------------------------------------
__global__ void __launch_bounds__(256)
dft64_cc(const float* __restrict__ ire, const float* __restrict__ iim,
         float* __restrict__ ore, float* __restrict__ oim,
         int stride, int tiles_per_outer, float sign)
{
  __shared__ float Fre[64 * 64];
  __shared__ float Fim[64 * 64];
  int tid = threadIdx.x;
  for (int i = tid; i < 4096; i += 256) {
    int k = i >> 6, m = i & 63;
    int km = (k * m) & 63;
    float s, c;
    sincosf(sign * 2.0f * PI_F * (float)km / 64.0f, &s, &c);
    Fre[i] = c; Fim[i] = s;
  }
  __syncthreads();

  int lane  = tid & 31;
  int tile  = blockIdx.x * 8 + (tid >> 5);
  int outer = tile / tiles_per_outer;
  int itile = tile - outer * tiles_per_outer;
  int base  = outer * 64 * stride + itile * 16 + (lane & 15);
  int khalf = (lane >> 4) << 1;
  int m = lane & 15;

  v8f zero = {};
  v8f accre[4] = {zero, zero, zero, zero};
  v8f accim[4] = {zero, zero, zero, zero};

#pragma unroll 4
  for (int kk = 0; kk < 16; ++kk) {
    int k0 = kk * 4 + khalf;
    v2f bre; bre.x = ire[base + k0 * stride]; bre.y = ire[base + (k0 + 1) * stride];
    v2f bim; bim.x = iim[base + k0 * stride]; bim.y = iim[base + (k0 + 1) * stride];
#pragma unroll
    for (int rt = 0; rt < 4; ++rt) {
      int mm = rt * 16 + m;
      v2f are;  are.x  = Fre[k0 * 64 + mm];  are.y  = Fre[(k0 + 1) * 64 + mm];
      v2f aim;  aim.x  = Fim[k0 * 64 + mm];  aim.y  = Fim[(k0 + 1) * 64 + mm];
      v2f aimn; aimn.x = -aim.x;             aimn.y = -aim.y;
      accre[rt] = wmma_f32(are,  bre, accre[rt]);
      accre[rt] = wmma_f32(aimn, bim, accre[rt]);
      accim[rt] = wmma_f32(are,  bim, accim[rt]);
      accim[rt] = wmma_f32(aim,  bre, accim[rt]);
    }
  }

  int rbase = (lane >> 4) * 8;
  int colb  = itile * 16 + (lane & 15);
#pragma unroll
  for (int rt = 0; rt < 4; ++rt) {
#pragma unroll
    for (int r = 0; r < 8; ++r) {
      int mm  = rt * 16 + rbase + r;
      int off = outer * 64 * stride + mm * stride + colb;
      ore[off] = accre[rt][r];
      oim[off] = accim[rt][r];
    }
  }
}

// ---------------------------------------------------------------------------
// Small prep kernels: A_stable, then separable kernel-DFT partials.
// P1[kt,kh,w,c] = sum_kw K[c,kt,kh,kw] * e^{-2pi i w (kw-3)/64}
// P2[kt,h ,w,c] = sum_kh P1 * e^{-2pi i h (kh-3)/64}
// ---------------------------------------------------------------------------
__global__ void k_astable(const float* __restrict__ A, float* __restrict__ As, int n)
{
  int i = blockIdx.x * 256 + threadIdx.x;
  if (i < n) As[i] = 0.9f * tanhf(A[i]);
}

__global__ void k_p1(const float* __restrict__ As, const float* __restrict__ Bk,
                     float2* __restrict__ P1A, float2* __restrict__ P1B)
{
  int i = blockIdx.x * 256 + threadIdx.x;   // ((kt*7+kh)*64+w)*128+c
  int c  = i & 127;
  int w  = (i >> 7) & 63;
  int kk = i >> 13;                          // kt*7+kh
  int kh = kk % 7;
  int kt = kk / 7;
  float2 aA = make_float2(0.f, 0.f), aB = make_float2(0.f, 0.f);
#pragma unroll
  for (int kw = 0; kw < 7; ++kw) {
    float s, cc;
    sincosf(-2.0f * PI_F * (float)(w * (kw - 3)) / 64.0f, &s, &cc);
    int ki = ((c * 3 + kt) * 7 + kh) * 7 + kw;   // kernels are (C,3,7,7)
    float a = As[ki], b = Bk[ki];
    aA.x += a * cc; aA.y += a * s;
    aB.x += b * cc; aB.y += b * s;
  }
  P1A[i] = aA; P1B[i] = aB;
}

__global__ void k_p2(const float2* __restrict__ P1A, const float2* __restrict__ P1B,
                     float2* __restrict__ P2A, float2* __restrict__ P2B)
{
  int i = blockIdx.x * 256 + threadIdx.x;   // ((kt*64+h)*64+w)*128+c
  int c  = i & 127;
  int w  = (i >> 7) & 63;
  int h  = (i >> 13) & 63;
  int kt = i >> 19;
  float2 aA = make_float2(0.f, 0.f), aB = make_float2(0.f, 0.f);
#pragma unroll
  for (int kh = 0; kh < 7; ++kh) {
    float s, cc;
    sincosf(-2.0f * PI_F * (float)(h * (kh - 3)) / 64.0f, &s, &cc);
    int pi = ((kt * 7 + kh) * 64 + w) * 128 + c;
    float2 pa = P1A[pi], pb = P1B[pi];
    aA.x += pa.x * cc - pa.y * s;  aA.y += pa.x * s + pa.y * cc;
    aB.x += pb.x * cc - pb.y * s;  aB.y += pb.x * s + pb.y * cc;
  }
  P2A[i] = aA; P2B[i] = aB;
}

// ---------------------------------------------------------------------------
// Pointwise: finish kt-sum for A_f/B_f, S = sum_{k=0..7} A_f^k (Horner),
// X *= B_f * S   (in place, streaming).
// ---------------------------------------------------------------------------
__global__ void k_pointwise(float* __restrict__ xre, float* __restrict__ xim,
                            const float2* __restrict__ P2A, const float2* __restrict__ P2B)
{
  int i = blockIdx.x * 256 + threadIdx.x;   // ((t*64+h)*64+w)*128+c
  int c = i & 127;
  int w = (i >> 7) & 63;
  int h = (i >> 13) & 63;
  int t = i >> 19;
  float2 Af = make_float2(0.f, 0.f), Bf = make_float2(0.f, 0.f);
#pragma unroll
  for (int kt = 0; kt < 3; ++kt) {
    float s, cc;
    sincosf(-2.0f * PI_F * (float)(t * (kt - 1)) / 16.0f, &s, &cc);
    int pi = ((kt * 64 + h) * 64 + w) * 128 + c;
    float2 pa = P2A[pi], pb = P2B[pi];
    Af.x += pa.x * cc - pa.y * s;  Af.y += pa.x * s + pa.y * cc;
    Bf.x += pb.x * cc - pb.y * s;  Bf.y += pb.x * s + pb.y * cc;
  }
  float2 S = make_float2(1.f, 0.f);
#pragma unroll
  for (int j = 0; j < 7; ++j) {
    float2 n;
    n.x = Af.x * S.x - Af.y * S.y + 1.0f;
    n.y = Af.x * S.y + Af.y * S.x;
    S = n;
  }
  float2 G;
  G.x = Bf.x * S.x - Bf.y * S.y;
  G.y = Bf.x * S.y + Bf.y * S.x;
  float2 X = make_float2(xre[i], xim[i]);
  xre[i] = G.x * X.x - G.y * X.y;
  xim[i] = G.x * X.y + G.y * X.x;
}

// ---------------------------------------------------------------------------
extern "C" void kernel_launch(void* const* d_in, const int* in_sizes, int n_in,
                              void* d_out, int out_size, void* d_ws, size_t ws_size,
                              hipStream_t stream)
{
  (void)in_sizes; (void)n_in; (void)out_size; (void)ws_size;
  const float* x  = (const float*)d_in[0];
  const float* Ak = (const float*)d_in[1];
  const float* Bk = (const float*)d_in[2];
  float* out = (float*)d_out;

  // Workspace carve (~162 MB total): ping-pong complex field + kernel partials.
  char* p = (char*)d_ws;
  float*  b0re = (float*)p;  p += (size_t)TOT * 4;
  float*  b0im = (float*)p;  p += (size_t)TOT * 4;
  float*  b1re = (float*)p;  p += (size_t)TOT * 4;
  float*  b1im = (float*)p;  p += (size_t)TOT * 4;
  float2* P1A  = (float2*)p; p += (size_t)(3 * 7 * 64 * 128) * 8;
  float2* P1B  = (float2*)p; p += (size_t)(3 * 7 * 64 * 128) * 8;
  float2* P2A  = (float2*)p; p += (size_t)(3 * 64 * 64 * 128) * 8;
  float2* P2B  = (float2*)p; p += (size_t)(3 * 64 * 64 * 128) * 8;
  float*  As   = (float*)p;  p += (size_t)(128 * 3 * 7 * 7) * 4;

  // Kernel-spectrum prep (cheap)
  k_astable<<<(128 * 147 + 255) / 256, 256, 0, stream>>>(Ak, As, 128 * 147);
  k_p1<<<(3 * 7 * 64 * 128) / 256, 256, 0, stream>>>(As, Bk, P1A, P1B);
  k_p2<<<(3 * 64 * 64 * 128) / 256, 256, 0, stream>>>(P1A, P1B, P2A, P2B);

  // Forward FFT3D(x) as WMMA GEMM stages: T(16), H(64), W(64)
  dft16_fwd_real<<<(STR_T / 16) / 8, 256, 0, stream>>>(x, b0re, b0im);
  dft64_cc<<<(TOT / (64 * 16)) / 8, 256, 0, stream>>>(b0re, b0im, b1re, b1im, STR_H, STR_H / 16, -1.0f);
  dft64_cc<<<(TOT / (64 * 16)) / 8, 256, 0, stream>>>(b1re, b1im, b0re, b0im, STR_W, STR_W / 16, -1.0f);

  // Frequency-domain transfer function (scan collapsed to geometric sum)
  k_pointwise<<<TOT / 256, 256, 0, stream>>>(b0re, b0im, P2A, P2B);

  // Inverse: W, H, then T (real part + 1/65536 scale) straight to d_out
  dft64_cc<<<(TOT / (64 * 16)) / 8, 256, 0, stream>>>(b0re, b0im, b1re, b1im, STR_W, STR_W / 16, 1.0f);
  dft64_cc<<<(TOT / (64 * 16)) / 8, 256, 0, stream>>>(b1re, b1im, b0re, b0im, STR_H, STR_H / 16, 1.0f);
  dft16_inv_real<<<(STR_T / 16) / 8, 256, 0, stream>>>(b0re, b0im, out);
}